// HyperedgeContrastiveLoss_48799418417343
// MI455X (gfx1250) — compile-verified
//
#include <hip/hip_runtime.h>
#include <hip/hip_bf16.h>
#include <math.h>

typedef __attribute__((ext_vector_type(2))) float v2f;
typedef __attribute__((ext_vector_type(8))) float v8f;
typedef int v4i __attribute__((ext_vector_type(4)));
typedef __attribute__((address_space(1))) v4i* as1_v4i_ptr;
typedef __attribute__((address_space(3))) v4i* as3_v4i_ptr;

#define D_DIM    256
#define TAU_INV  10.0f
#define EPS_F    1e-8f
#define COLSPLIT 4

#if __has_builtin(__builtin_amdgcn_global_load_async_to_lds_b128) && \
    __has_builtin(__builtin_amdgcn_s_wait_asynccnt)
#define HAS_ASYNC_LDS 1
#else
#define HAS_ASYNC_LDS 0
#endif

// ---------------------------------------------------------------------------
// Kernel 0: edge start offsets via binary search (member_edge_id is sorted)
// ---------------------------------------------------------------------------
__global__ void k_offsets(const int* __restrict__ member_edge_id, int M, int E,
                          int* __restrict__ off) {
  int e = blockIdx.x * blockDim.x + threadIdx.x;
  if (e > E) return;
  if (e == E) { off[e] = M; return; }
  int lo = 0, hi = M;
  while (lo < hi) {
    int mid = (lo + hi) >> 1;
    if (member_edge_id[mid] < e) lo = mid + 1; else hi = mid;
  }
  off[e] = lo;
}

// ---------------------------------------------------------------------------
// Kernel 1: per-edge mean, L2-normalized embedding, fused intra term
//   intra_e = (sum_d ss_d - cnt * sum_d mean_d^2) / cnt
// ---------------------------------------------------------------------------
__global__ void k_edge_stats(const float* __restrict__ z_nodes,
                             const int*   __restrict__ member_node_idx,
                             const int*   __restrict__ off,
                             float* __restrict__ edge_emb,
                             float* __restrict__ znorm,
                             float* __restrict__ per_edge_intra) {
  __shared__ float red_m2[256];
  __shared__ float red_ss[256];
  const int e = blockIdx.x;
  const int d = threadIdx.x;
  const int start = off[e], end = off[e + 1];
  const float cnt = (float)(end - start);
  float s = 0.f, ss = 0.f;
  for (int m = start; m < end; ++m) {
    int node = member_node_idx[m];
    float x = z_nodes[(size_t)node * D_DIM + d];
    s += x; ss += x * x;
  }
  const float mean = s / cnt;
  edge_emb[(size_t)e * D_DIM + d] = mean;
  red_m2[d] = mean * mean;
  red_ss[d] = ss;
  __syncthreads();
  for (int st = 128; st > 0; st >>= 1) {
    if (d < st) { red_m2[d] += red_m2[d + st]; red_ss[d] += red_ss[d + st]; }
    __syncthreads();
  }
  const float sum_m2 = red_m2[0];
  const float sum_ss = red_ss[0];
  znorm[(size_t)e * D_DIM + d] = mean * rsqrtf(sum_m2);
  if (d == 0) per_edge_intra[e] = (sum_ss - cnt * sum_m2) / cnt;
}

// ---------------------------------------------------------------------------
// B-tile-pair copy: 32 cols x 256 K (32 KB) global -> LDS.
// LDS layout (memory order within 4-float groups, rotation-swizzled on n):
//   float index = p*4096 + g*64 + ((n+g)&15)*4 + (k&3),  g = k>>2, p = tile.
// Async path: one GLOBAL_LOAD_ASYNC_TO_LDS_B128 per 16B (ASYNCcnt-tracked).
// ---------------------------------------------------------------------------
__device__ __forceinline__ void copy_pair(const float* __restrict__ znorm,
                                          float* lds_dst, int col0, int tid) {
#pragma unroll
  for (int it = 0; it < 16; ++it) {
    int c  = it * 128 + tid;          // 2048 x 16B copies, 128 threads
    int p  = c >> 10;
    int q  = c & 1023;
    int nn = q >> 6;
    int g  = q & 63;
    const float* gsrc = znorm + (size_t)(col0 + p * 16 + nn) * D_DIM + g * 4;
    float* ldst = lds_dst + p * 4096 + g * 64 + (((nn + g) & 15) << 2);
#if HAS_ASYNC_LDS
    __builtin_amdgcn_global_load_async_to_lds_b128(
        (as1_v4i_ptr)(uintptr_t)gsrc,
        (as3_v4i_ptr)(uintptr_t)ldst, 0, 0);
#else
    *(float4*)ldst = *(const float4*)gsrc;
#endif
  }
}

__device__ __forceinline__ void wait_copies() {
#if HAS_ASYNC_LDS
  __builtin_amdgcn_s_wait_asynccnt(0);
#endif
}

// ---------------------------------------------------------------------------
// Kernel 2: fused  sim = (z z^T)/tau -> exp -> masked row sums.
// fp32 WMMA (v_wmma_f32_16x16x4_f32). 4 waves/WG, each wave owns a 16-row
// block (A panel in registers). Two column tiles per iteration -> two
// independent WMMA accumulation chains (breaks the C-RAW stall). Double-
// buffered 2x32KB dynamic LDS; next pair fetched asynchronously while the
// current pair is consumed. Column range split COLSPLIT ways across
// blockIdx.y; per-split partial sums (single writer -> deterministic).
// ---------------------------------------------------------------------------
__global__ void __launch_bounds__(128)
k_sim_rowsums(const float* __restrict__ znorm,
              const int*   __restrict__ edge_type,
              int E,
              float* __restrict__ denomPart,   // [COLSPLIT][E]
              float* __restrict__ numerPart) { // [COLSPLIT][E]
  extern __shared__ float ldsB[];              // 2 * 8192 floats = 64 KB
  const int tid  = threadIdx.x;
  const int lane = tid & 31;
  const int wid  = tid >> 5;
  const int n    = lane & 15;                  // tile column / A row
  const int hi   = lane >> 4;                  // half-wave select
  const int i0   = (blockIdx.x * 4 + wid) * 16;
  const int trow = edge_type[i0];

  // A panel: 16 rows x 256 K.  ISA: lanes0-15 K={kk,kk+1}, lanes16-31 K={kk+2,kk+3}
  v2f a[64];
  {
    const float* arow = znorm + (size_t)(i0 + n) * D_DIM;
#pragma unroll
    for (int t = 0; t < 64; ++t)
      a[t] = *(const v2f*)(arow + 4 * t + 2 * hi);
  }

  float dAcc[8], nAcc[8];
#pragma unroll
  for (int v = 0; v < 8; ++v) { dAcc[v] = 0.f; nAcc[v] = 0.f; }

  const int nJ     = E >> 4;
  const int perWG  = nJ / COLSPLIT;            // 128 tiles = 64 pairs
  const int jFirst = blockIdx.y * perWG;
  const int jLast  = jFirst + perWG;

  int cur = 0;
  copy_pair(znorm, ldsB, jFirst << 4, tid);

  for (int jp = jFirst; jp < jLast; jp += 2) {
    wait_copies();
    __syncthreads();                           // buf[cur] ready; buf[cur^1] free
    if (jp + 2 < jLast)
      copy_pair(znorm, ldsB + (cur ^ 1) * 8192, (jp + 2) << 4, tid);

    const float* bufc = ldsB + cur * 8192;
    const int j0 = jp << 4;

    v8f c0 = {0.f,0.f,0.f,0.f,0.f,0.f,0.f,0.f};
    v8f c1 = {0.f,0.f,0.f,0.f,0.f,0.f,0.f,0.f};
#pragma unroll
    for (int t = 0; t < 64; ++t) {
      const int boff = t * 64 + (((n + t) & 15) << 2) + hi;
      v2f b0, b1;
      b0.x = bufc[boff];            b0.y = bufc[boff + 2];
      b1.x = bufc[4096 + boff];     b1.y = bufc[4096 + boff + 2];
      c0 = __builtin_amdgcn_wmma_f32_16x16x4_f32(false, a[t], false, b0,
                                                 (short)0, c0, false, false);
      c1 = __builtin_amdgcn_wmma_f32_16x16x4_f32(false, a[t], false, b1,
                                                 (short)0, c1, false, false);
    }

    const bool tm0 = (edge_type[j0]      == trow);
    const bool tm1 = (edge_type[j0 + 16] == trow);
    const int col0 = j0 + n;
    const int col1 = j0 + 16 + n;
#pragma unroll
    for (int v = 0; v < 8; ++v) {
      const int row = i0 + v + 8 * hi;         // C/D layout: vgpr v, half hi
      float e0 = __expf(c0[v] * TAU_INV);
      float e1 = __expf(c1[v] * TAU_INV);
      e0 = (row == col0) ? 0.f : e0;
      e1 = (row == col1) ? 0.f : e1;
      dAcc[v] += e0 + e1;
      nAcc[v] += (tm0 ? e0 : 0.f) + (tm1 ? e1 : 0.f);
    }
    cur ^= 1;
  }

  // Row sums live across the 16 lanes of each half-wave -> butterfly reduce.
  float* dOut = denomPart + (size_t)blockIdx.y * E;
  float* nOut = numerPart + (size_t)blockIdx.y * E;
#pragma unroll
  for (int v = 0; v < 8; ++v) {
    float dv = dAcc[v], nv = nAcc[v];
    for (int o = 8; o > 0; o >>= 1) {
      dv += __shfl_xor(dv, o, 16);
      nv += __shfl_xor(nv, o, 16);
    }
    if (n == 0) {
      const int row = i0 + v + 8 * hi;
      dOut[row] = dv;
      nOut[row] = nv;
    }
  }
}

// ---------------------------------------------------------------------------
// Kernel 3: finalize all three losses + total (single block)
// ---------------------------------------------------------------------------
__global__ void k_finalize(const float* __restrict__ per_edge_intra,
                           const float* __restrict__ denomPart,
                           const float* __restrict__ numerPart,
                           const float* __restrict__ edge_emb,
                           const int*   __restrict__ intra_sample, int n_intra,
                           const int*   __restrict__ pair1,
                           const int*   __restrict__ pair2, int n_pairs,
                           int E, float* __restrict__ out) {
  __shared__ float rT[256], rI[256], rP[256];
  const int t = threadIdx.x;
  float sT = 0.f, sI = 0.f, sP = 0.f;
  for (int e = t; e < E; e += 256) {                    // InfoNCE rows
    float den = 0.f, num = 0.f;
#pragma unroll
    for (int s = 0; s < COLSPLIT; ++s) {
      den += denomPart[(size_t)s * E + e];
      num += numerPart[(size_t)s * E + e];
    }
    sT += -logf((num + EPS_F) / (den + EPS_F));
  }
  for (int i = t; i < n_intra; i += 256)                // sampled intra edges
    sI += per_edge_intra[intra_sample[i]];
  for (int p = t; p < n_pairs; p += 256) {              // margin pairs
    const float* e1 = edge_emb + (size_t)pair1[p] * D_DIM;
    const float* e2 = edge_emb + (size_t)pair2[p] * D_DIM;
    float d2 = 0.f;
    for (int d = 0; d < D_DIM; ++d) { float df = e1[d] - e2[d]; d2 += df * df; }
    float r = fmaxf(1.0f - sqrtf(d2), 0.f);
    sP += r * r;
  }
  rT[t] = sT; rI[t] = sI; rP[t] = sP;
  __syncthreads();
  for (int st = 128; st > 0; st >>= 1) {
    if (t < st) { rT[t] += rT[t + st]; rI[t] += rI[t + st]; rP[t] += rP[t + st]; }
    __syncthreads();
  }
  if (t == 0) {
    float intra = rI[0] / (float)n_intra;
    float tloss = rT[0] / (float)E;          // every row has 2047 positives
    float inter = rP[0] / (float)n_pairs;
    out[0] = intra;
    out[1] = tloss;
    out[2] = inter;
    out[3] = 1.0f * intra + 0.5f * tloss + 0.5f * inter;
  }
}

// ---------------------------------------------------------------------------
extern "C" void kernel_launch(void* const* d_in, const int* in_sizes, int n_in,
                              void* d_out, int out_size, void* d_ws, size_t ws_size,
                              hipStream_t stream) {
  const float* z_nodes         = (const float*)d_in[0];
  const int*   member_node_idx = (const int*)d_in[1];
  const int*   member_edge_id  = (const int*)d_in[2];
  const int*   edge_type       = (const int*)d_in[3];
  const int*   intra_sample    = (const int*)d_in[4];
  const int*   pair_idx1       = (const int*)d_in[5];
  const int*   pair_idx2       = (const int*)d_in[6];
  float* out = (float*)d_out;

  const int M = in_sizes[1];
  const int E = in_sizes[3];
  const int n_intra = in_sizes[4];
  const int n_pairs = in_sizes[5];

  // workspace layout (all regions fully rewritten each call)
  char* base = (char*)d_ws;
  size_t cur = 0;
  auto carve = [&](size_t bytes) { void* p = base + cur;
                                   cur = (cur + bytes + 255) & ~(size_t)255; return p; };
  int*   off            = (int*)  carve((size_t)(E + 1) * 4);
  float* per_edge_intra = (float*)carve((size_t)E * 4);
  float* denomPart      = (float*)carve((size_t)COLSPLIT * E * 4);
  float* numerPart      = (float*)carve((size_t)COLSPLIT * E * 4);
  float* edge_emb       = (float*)carve((size_t)E * D_DIM * 4);
  float* znorm          = (float*)carve((size_t)E * D_DIM * 4);

  k_offsets<<<(E + 1 + 255) / 256, 256, 0, stream>>>(member_edge_id, M, E, off);
  k_edge_stats<<<E, 256, 0, stream>>>(z_nodes, member_node_idx, off,
                                      edge_emb, znorm, per_edge_intra);
  dim3 gridSim(E / 64, COLSPLIT);
  k_sim_rowsums<<<gridSim, 128, 2 * 8192 * sizeof(float), stream>>>(
      znorm, edge_type, E, denomPart, numerPart);
  k_finalize<<<1, 256, 0, stream>>>(per_edge_intra, denomPart, numerPart,
                                    edge_emb, intra_sample, n_intra,
                                    pair_idx1, pair_idx2, n_pairs, E, out);
}